// RecGCNblock_37838661877767
// MI455X (gfx1250) — compile-verified
//
#include <hip/hip_runtime.h>
#include <math.h>

typedef __attribute__((ext_vector_type(2))) float v2f;
typedef __attribute__((ext_vector_type(8))) float v8f;

#define DIM 128

static __device__ __forceinline__ v8f wmma4(v2f a, v2f b, v8f c) {
    // V_WMMA_F32_16X16X4_F32 : D = A(16x4) * B(4x16) + C(16x16), fp32
    return __builtin_amdgcn_wmma_f32_16x16x4_f32(
        /*neg_a=*/false, a, /*neg_b=*/false, b,
        /*c_mod=*/(short)0, c, /*reuse_a=*/false, /*reuse_b=*/false);
}

// ---------------------------------------------------------------- zero
__global__ void k_zero(float* p, long n) {
    long i = (long)blockIdx.x * blockDim.x + threadIdx.x;
    long stride = (long)gridDim.x * blockDim.x;
    for (; i < n; i += stride) p[i] = 0.0f;
}

// ---------------------------------------------------------------- degrees
__global__ void k_degree(const int* __restrict__ src, const int* __restrict__ dst,
                         float* deg_out, float* deg_in, int nE) {
    int e = blockIdx.x * blockDim.x + threadIdx.x;
    if (e < nE) {
        atomicAdd(&deg_out[src[e]], 1.0f);
        atomicAdd(&deg_in[dst[e]], 1.0f);
    }
}

// ---------------------------------------------------------------- norms
__global__ void k_norm(const float* __restrict__ deg_out, const float* __restrict__ deg_in,
                       float* nsrc, float* ndst, int nN) {
    int i = blockIdx.x * blockDim.x + threadIdx.x;
    if (i < nN) {
        float a = deg_out[i];
        float b = deg_in[i];
        nsrc[i] = (a > 0.0f) ? rsqrtf(fmaxf(a, 1.0f)) : 0.0f;
        ndst[i] = (b > 0.0f) ? rsqrtf(fmaxf(b, 1.0f)) : 0.0f;
    }
}

// ---------------------------------------------------------------- edge scatter
// one wave (32 lanes) per edge; lane handles 4 consecutive floats (float4 gather)
__global__ __launch_bounds__(256) void k_scatter(const int* __restrict__ src,
                                                 const int* __restrict__ dst,
                                                 const float* __restrict__ h,
                                                 const float* __restrict__ nsrc,
                                                 float* agg, int nE) {
    int lane = threadIdx.x & 31;
    int wv   = threadIdx.x >> 5;
    int e = blockIdx.x * 8 + wv;
    if (e >= nE) return;
    int s = src[e];
    int d = dst[e];
    float sc = nsrc[s];
    float4 v = ((const float4*)(h + (long)s * DIM))[lane];
    float* ap = agg + (long)d * DIM + lane * 4;
    atomicAdd(ap + 0, v.x * sc);
    atomicAdd(ap + 1, v.y * sc);
    atomicAdd(ap + 2, v.z * sc);
    atomicAdd(ap + 3, v.w * sc);
}

// ---------------------------------------------------------------- GCN GEMM
// x2 = (agg * ndst[:,None]) @ W + b    via V_WMMA_F32_16X16X4_F32
// 8 waves / WG, one 16-row tile per wave, N=128 split into 8 col-tiles.
__global__ __launch_bounds__(256) void k_gcn_gemm(const float* __restrict__ agg,
                                                  const float* __restrict__ ndst,
                                                  const float* __restrict__ W,
                                                  const float* __restrict__ bias,
                                                  float* x2, int nN, int nTiles) {
    __shared__ float lds_a[8][16][DIM];           // 64 KB
    int lane = threadIdx.x & 31;
    int wv   = threadIdx.x >> 5;
    int t = blockIdx.x * 8 + wv;
    if (t >= nTiles) return;
    int m0 = t * 16;

    // stage A tile (scaled by norm_dst), coalesced float4, per-wave region
    for (int r = 0; r < 16; ++r) {
        int row = m0 + r; if (row >= nN) row = nN - 1;
        float sc = ndst[row];
        float4 v = ((const float4*)(agg + (long)row * DIM))[lane];
        v.x *= sc; v.y *= sc; v.z *= sc; v.w *= sc;
        ((float4*)&lds_a[wv][r][0])[lane] = v;
    }

    v8f acc[8];
#pragma unroll
    for (int n = 0; n < 8; ++n) acc[n] = (v8f){};

    int mrow  = lane & 15;
    int khalf = (lane >> 4) << 1;

    for (int k0 = 0; k0 < DIM; k0 += 4) {
        int kk = k0 + khalf;
        v2f a = *(const v2f*)&lds_a[wv][mrow][kk];      // ds_load_b64
#pragma unroll
        for (int n = 0; n < 8; ++n) {
            int col = n * 16 + mrow;
            v2f b;
            b.x = W[(long)kk * DIM + col];              // W[k][n]
            b.y = W[(long)(kk + 1) * DIM + col];
            acc[n] = wmma4(a, b, acc[n]);
        }
    }

#pragma unroll
    for (int n = 0; n < 8; ++n) {
        int col = n * 16 + mrow;
        float bv = bias[col];
#pragma unroll
        for (int i = 0; i < 8; ++i) {
            int M = i + ((lane >> 4) << 3);             // C layout: row = i + 8*(lane>=16)
            int row = m0 + M;
            if (row < nN) x2[(long)row * DIM + col] = acc[n][i] + bv;
        }
    }
}

// ---------------------------------------------------------------- fused GRU + ELU
// gi = x2 @ w_ih^T + b_ih ; gh = h @ w_hh^T + b_hh ; gates ; elu
__global__ __launch_bounds__(256) void k_gru(const float* __restrict__ x2,
                                             const float* __restrict__ h,
                                             const float* __restrict__ w_ih,
                                             const float* __restrict__ w_hh,
                                             const float* __restrict__ b_ih,
                                             const float* __restrict__ b_hh,
                                             float* out, int nN, int nTiles) {
    __shared__ float lx[8][16][DIM];                  // 64 KB
    __shared__ float lh[8][16][DIM];                  // 64 KB
    int lane = threadIdx.x & 31;
    int wv   = threadIdx.x >> 5;
    int t = blockIdx.x * 8 + wv;
    if (t >= nTiles) return;
    int m0 = t * 16;

    for (int r = 0; r < 16; ++r) {
        int row = m0 + r; if (row >= nN) row = nN - 1;
        ((float4*)&lx[wv][r][0])[lane] = ((const float4*)(x2 + (long)row * DIM))[lane];
        ((float4*)&lh[wv][r][0])[lane] = ((const float4*)(h  + (long)row * DIM))[lane];
    }

    int mrow  = lane & 15;
    int khalf = (lane >> 4) << 1;

    for (int j = 0; j < 8; ++j) {                     // output column tiles
        int col = j * 16 + mrow;
        v8f a_ir = (v8f){}, a_iz = (v8f){}, a_in = (v8f){};
        v8f a_hr = (v8f){}, a_hz = (v8f){}, a_hn = (v8f){};

        // w_ih/w_hh are (3*DIM, DIM) row-major; B[k][n] = w[n][k] -> K-consecutive b64 loads
        const float* wr = w_ih + (long)(0 * DIM + col) * DIM;
        const float* wz = w_ih + (long)(1 * DIM + col) * DIM;
        const float* wn = w_ih + (long)(2 * DIM + col) * DIM;
        const float* ur = w_hh + (long)(0 * DIM + col) * DIM;
        const float* uz = w_hh + (long)(1 * DIM + col) * DIM;
        const float* un = w_hh + (long)(2 * DIM + col) * DIM;

        for (int k0 = 0; k0 < DIM; k0 += 4) {
            int kk = k0 + khalf;
            v2f ax = *(const v2f*)&lx[wv][mrow][kk];
            v2f ah = *(const v2f*)&lh[wv][mrow][kk];
            v2f br = *(const v2f*)(wr + kk);
            v2f bz = *(const v2f*)(wz + kk);
            v2f bn = *(const v2f*)(wn + kk);
            v2f cr = *(const v2f*)(ur + kk);
            v2f cz = *(const v2f*)(uz + kk);
            v2f cn = *(const v2f*)(un + kk);
            a_ir = wmma4(ax, br, a_ir);
            a_iz = wmma4(ax, bz, a_iz);
            a_in = wmma4(ax, bn, a_in);
            a_hr = wmma4(ah, cr, a_hr);
            a_hz = wmma4(ah, cz, a_hz);
            a_hn = wmma4(ah, cn, a_hn);
        }

        float bir = b_ih[col], biz = b_ih[DIM + col], bin = b_ih[2 * DIM + col];
        float bhr = b_hh[col], bhz = b_hh[DIM + col], bhn = b_hh[2 * DIM + col];

#pragma unroll
        for (int i = 0; i < 8; ++i) {
            int M = i + ((lane >> 4) << 3);
            float r  = 1.0f / (1.0f + __expf(-(a_ir[i] + bir + a_hr[i] + bhr)));
            float z  = 1.0f / (1.0f + __expf(-(a_iz[i] + biz + a_hz[i] + bhz)));
            float nn = tanhf(a_in[i] + bin + r * (a_hn[i] + bhn));
            float hv = lh[wv][M][col];
            float hnew = (1.0f - z) * nn + z * hv;
            float o = (hnew > 0.0f) ? hnew : (__expf(hnew) - 1.0f);   // ELU, alpha=1
            int row = m0 + M;
            if (row < nN) out[(long)row * DIM + col] = o;
        }
    }
}

// ---------------------------------------------------------------- launch
extern "C" void kernel_launch(void* const* d_in, const int* in_sizes, int n_in,
                              void* d_out, int out_size, void* d_ws, size_t ws_size,
                              hipStream_t stream) {
    const int*   edge_src = (const int*)d_in[0];
    const int*   edge_dst = (const int*)d_in[1];
    const float* h        = (const float*)d_in[2];
    const float* gcn_W    = (const float*)d_in[3];
    const float* gcn_b    = (const float*)d_in[4];
    const float* w_ih     = (const float*)d_in[5];
    const float* w_hh     = (const float*)d_in[6];
    const float* b_ih     = (const float*)d_in[7];
    const float* b_hh     = (const float*)d_in[8];

    const int nE = in_sizes[0];
    const int nN = in_sizes[2] / DIM;
    const int nTiles = (nN + 15) / 16;

    float* ws      = (float*)d_ws;
    float* deg_out = ws;                      // nN
    float* deg_in  = deg_out + nN;            // nN
    float* nsrc    = deg_in + nN;             // nN
    float* ndst    = nsrc + nN;               // nN
    float* agg     = ndst + nN;               // nN * DIM
    float* x2      = agg + (long)nN * DIM;    // nN * DIM
    float* outp    = (float*)d_out;

    // zero degs/norms/agg in one grid-stride pass
    long nzero = (long)nN * (4 + DIM);
    k_zero<<<2048, 256, 0, stream>>>(ws, nzero);

    k_degree<<<(nE + 255) / 256, 256, 0, stream>>>(edge_src, edge_dst, deg_out, deg_in, nE);
    k_norm<<<(nN + 255) / 256, 256, 0, stream>>>(deg_out, deg_in, nsrc, ndst, nN);
    k_scatter<<<(nE + 7) / 8, 256, 0, stream>>>(edge_src, edge_dst, h, nsrc, agg, nE);
    k_gcn_gemm<<<(nTiles + 7) / 8, 256, 0, stream>>>(agg, ndst, gcn_W, gcn_b, x2, nN, nTiles);
    k_gru<<<(nTiles + 7) / 8, 256, 0, stream>>>(x2, h, w_ih, w_hh, b_ih, b_hh, outp, nN, nTiles);
}